// HardClusterAssigner_54735063220662
// MI455X (gfx1250) — compile-verified
//
#include <hip/hip_runtime.h>

// HardClusterAssigner for MI455X (gfx1250), wave32 + WMMA f32 path.
//
// Pipeline (all scratch in d_ws, fully overwritten each call):
//   K1: xm[v,s] = mean_b x[b,s,v]            (memory-bound: 134 MB NT stream)
//   K2: E = xm * W^T + b   via V_WMMA_F32_16X16X4_F32   (512x512x1024)
//   K3: rn[c] = 1 / max(||cent[c]||, 1e-12)
//   K4: G = E * cent^T     via V_WMMA_F32_16X16X4_F32   (512x64x512)
//   K5: argmax_c G[v,c]*rn[c]  ->  one-hot rows of d_out
//
// Row-normalization of E is folded out (positive per-row scalar cannot
// change the argmin), so full f32 precision is preserved end-to-end.
// GEMM k-loop is software-pipelined (prefetch distance 2) so s_wait_loadcnt
// before each WMMA group waits on loads issued 2 iterations earlier.

typedef __attribute__((ext_vector_type(2))) float v2f;
typedef __attribute__((ext_vector_type(4))) float v4f;
typedef __attribute__((ext_vector_type(8))) float v8f;

constexpr int Bc = 64;    // batch
constexpr int Sc = 1024;  // seq len (contraction dim of linear)
constexpr int Vc = 512;   // vocab rows
constexpr int Hc = 512;   // hidden
constexpr int Cc = 64;    // clusters

// ---------------------------------------------------------------------------
// K1: mean over batch with transpose.  x:[B,S,V] -> xm:[V,S]
// Consecutive threads read consecutive v (b128) -> fully coalesced 134 MB
// stream. Non-temporal: x is single-use; keep L2 for xm/W/E working set.
// ---------------------------------------------------------------------------
__global__ void mean_bt_kernel(const float* __restrict__ x,
                               float* __restrict__ xm) {
  const int tid = blockIdx.x * blockDim.x + threadIdx.x;  // 0 .. S*V/4-1
  const int v4 = tid % (Vc / 4);
  const int s  = tid / (Vc / 4);
  const v4f* xp = reinterpret_cast<const v4f*>(x) + (size_t)s * (Vc / 4) + v4;
  v4f acc = {0.f, 0.f, 0.f, 0.f};
#pragma unroll 4
  for (int b = 0; b < Bc; ++b) {
    v4f t = __builtin_nontemporal_load(xp + (size_t)b * (Sc * (Vc / 4)));
    acc += t;
  }
  const float inv = 1.0f / (float)Bc;
  const int v = v4 * 4;
  xm[(size_t)(v + 0) * Sc + s] = acc.x * inv;
  xm[(size_t)(v + 1) * Sc + s] = acc.y * inv;
  xm[(size_t)(v + 2) * Sc + s] = acc.z * inv;
  xm[(size_t)(v + 3) * Sc + s] = acc.w * inv;
}

// ---------------------------------------------------------------------------
// K2/K4: C[m,n] = sum_k A[m,k]*Bm[n,k] (+ bias[n])  — NT GEMM, f32 WMMA.
// One wave per 16(M) x 64(N) tile; A operand reused across 4 WMMAs per k-step.
//
// V_WMMA_F32_16X16X4_F32 operand layout (ISA 7.12.2, 32-bit A 16x4):
//   A: lane L (L<16): M=L, vgpr{0,1}=K{0,1}; lane L>=16: M=L-16, vgpr{0,1}=K{2,3}
//   B (4x16, K x N): mirrored over N: lane L: N=L&15, vgpr{0,1}=K{kh,kh+1},
//      kh = 2*(L>>4)
//   C/D: vgpr r: lanes 0-15 -> M=r, lanes 16-31 -> M=r+8; N = lane&15
// ---------------------------------------------------------------------------
#define WMMA_F32(A_, B_, C_)                                                   \
  __builtin_amdgcn_wmma_f32_16x16x4_f32(false, (A_), false, (B_), (short)0,    \
                                        (C_), false, false)

__global__ void wmma_gemm_nt_kernel(const float* __restrict__ A,
                                    const float* __restrict__ Bm,
                                    const float* __restrict__ bias,
                                    float* __restrict__ Cout,
                                    int K, int ldc) {
  const int lane  = threadIdx.x;        // 0..31, EXEC all ones
  const int m0    = blockIdx.x * 16;
  const int n0    = blockIdx.y * 64;
  const int row   = lane & 15;
  const int kh    = (lane >> 4) * 2;    // 0 or 2

  const float* __restrict__ Ap  = A  + (size_t)(m0 + row) * K + kh;
  const float* __restrict__ Bp0 = Bm + (size_t)(n0 +  0 + row) * K + kh;
  const float* __restrict__ Bp1 = Bm + (size_t)(n0 + 16 + row) * K + kh;
  const float* __restrict__ Bp2 = Bm + (size_t)(n0 + 32 + row) * K + kh;
  const float* __restrict__ Bp3 = Bm + (size_t)(n0 + 48 + row) * K + kh;

  v8f acc0 = {}, acc1 = {}, acc2 = {}, acc3 = {};

  // --- software pipeline: groups of k=4; prefetch distance 2 ---------------
  v2f a_c  = *(const v2f*)(Ap);
  v2f b0_c = *(const v2f*)(Bp0);
  v2f b1_c = *(const v2f*)(Bp1);
  v2f b2_c = *(const v2f*)(Bp2);
  v2f b3_c = *(const v2f*)(Bp3);
  v2f a_n  = *(const v2f*)(Ap  + 4);
  v2f b0_n = *(const v2f*)(Bp0 + 4);
  v2f b1_n = *(const v2f*)(Bp1 + 4);
  v2f b2_n = *(const v2f*)(Bp2 + 4);
  v2f b3_n = *(const v2f*)(Bp3 + 4);

#pragma unroll 2
  for (int k = 0; k < K - 8; k += 4) {
    // prefetch group k+8 while computing on group k
    v2f a_p  = *(const v2f*)(Ap  + k + 8);
    v2f b0_p = *(const v2f*)(Bp0 + k + 8);
    v2f b1_p = *(const v2f*)(Bp1 + k + 8);
    v2f b2_p = *(const v2f*)(Bp2 + k + 8);
    v2f b3_p = *(const v2f*)(Bp3 + k + 8);

    acc0 = WMMA_F32(a_c, b0_c, acc0);
    acc1 = WMMA_F32(a_c, b1_c, acc1);
    acc2 = WMMA_F32(a_c, b2_c, acc2);
    acc3 = WMMA_F32(a_c, b3_c, acc3);

    a_c = a_n;  b0_c = b0_n; b1_c = b1_n; b2_c = b2_n; b3_c = b3_n;
    a_n = a_p;  b0_n = b0_p; b1_n = b1_p; b2_n = b2_p; b3_n = b3_p;
  }

  // epilogue: drain the two in-flight groups
  acc0 = WMMA_F32(a_c, b0_c, acc0);
  acc1 = WMMA_F32(a_c, b1_c, acc1);
  acc2 = WMMA_F32(a_c, b2_c, acc2);
  acc3 = WMMA_F32(a_c, b3_c, acc3);
  acc0 = WMMA_F32(a_n, b0_n, acc0);
  acc1 = WMMA_F32(a_n, b1_n, acc1);
  acc2 = WMMA_F32(a_n, b2_n, acc2);
  acc3 = WMMA_F32(a_n, b3_n, acc3);

  const int nlane = lane & 15;
  const int mhalf = (lane >> 4) * 8;
  const float bv0 = bias ? bias[n0 +  0 + nlane] : 0.0f;
  const float bv1 = bias ? bias[n0 + 16 + nlane] : 0.0f;
  const float bv2 = bias ? bias[n0 + 32 + nlane] : 0.0f;
  const float bv3 = bias ? bias[n0 + 48 + nlane] : 0.0f;
#pragma unroll
  for (int r = 0; r < 8; ++r) {
    const size_t m = (size_t)(m0 + mhalf + r);
    Cout[m * ldc + n0 +  0 + nlane] = acc0[r] + bv0;
    Cout[m * ldc + n0 + 16 + nlane] = acc1[r] + bv1;
    Cout[m * ldc + n0 + 32 + nlane] = acc2[r] + bv2;
    Cout[m * ldc + n0 + 48 + nlane] = acc3[r] + bv3;
  }
}

// ---------------------------------------------------------------------------
// K3: reciprocal centroid norms with the F.normalize eps clamp.
// ---------------------------------------------------------------------------
__global__ void cnorm_kernel(const float* __restrict__ cent,
                             float* __restrict__ rn) {
  const int c = threadIdx.x;  // one block of Cc threads
  float s = 0.0f;
  for (int h = 0; h < Hc; ++h) {
    const float t = cent[(size_t)c * Hc + h];
    s += t * t;
  }
  const float n = fmaxf(sqrtf(s), 1e-12f);
  rn[c] = 1.0f / n;
}

// ---------------------------------------------------------------------------
// K5: per-v first-argmax of G[v,c]*rn[c]  (== argmin of -cosine sim; the
// positive per-row 1/||E[v]|| factor is dropped as argmin-invariant).
// Writes the full one-hot output (d_out is poisoned, so every slot written).
// ---------------------------------------------------------------------------
__global__ void argmax_onehot_kernel(const float* __restrict__ G,
                                     const float* __restrict__ rn,
                                     float* __restrict__ out) {
  const int v = blockIdx.x * blockDim.x + threadIdx.x;  // 0..Vc-1
  float best = -__builtin_inff();
  int bi = 0;
#pragma unroll
  for (int c = 0; c < Cc; ++c) {
    const float val = G[(size_t)v * Cc + c] * rn[c];
    if (val > best) { best = val; bi = c; }  // strict > keeps FIRST max
  }
#pragma unroll
  for (int c = 0; c < Cc; ++c) {
    out[(size_t)v * Cc + c] = (c == bi) ? 1.0f : 0.0f;
  }
}

// ---------------------------------------------------------------------------
extern "C" void kernel_launch(void* const* d_in, const int* in_sizes, int n_in,
                              void* d_out, int out_size, void* d_ws,
                              size_t ws_size, hipStream_t stream) {
  const float* x    = (const float*)d_in[0];  // [B,S,V]
  const float* W    = (const float*)d_in[1];  // [H,S]
  const float* bias = (const float*)d_in[2];  // [H]
  const float* cent = (const float*)d_in[3];  // [C,H]
  float* out = (float*)d_out;                 // [V,C]

  float* xm = (float*)d_ws;                   // [V,S]   2 MB
  float* E  = xm + (size_t)Vc * Sc;           // [V,H]   1 MB
  float* G  = E + (size_t)Vc * Hc;            // [V,C]   128 KB
  float* rn = G + (size_t)Vc * Cc;            // [C]

  // K1: batch-mean + transpose (dominant: one 134 MB stream ~5.8 us @23.3TB/s)
  mean_bt_kernel<<<dim3((Sc * Vc / 4) / 256), dim3(256), 0, stream>>>(x, xm);

  // K2: E = xm * W^T + b   (M=512, N=512, K=1024)
  wmma_gemm_nt_kernel<<<dim3(Vc / 16, Hc / 64), dim3(32), 0, stream>>>(
      xm, W, bias, E, Sc, Hc);

  // K3: centroid reciprocal norms
  cnorm_kernel<<<dim3(1), dim3(Cc), 0, stream>>>(cent, rn);

  // K4: G = E * cent^T     (M=512, N=64, K=512)
  wmma_gemm_nt_kernel<<<dim3(Vc / 16, Cc / 64), dim3(32), 0, stream>>>(
      E, cent, nullptr, G, Hc, Cc);

  // K5: argmax + one-hot
  argmax_onehot_kernel<<<dim3(Vc / 256), dim3(256), 0, stream>>>(G, rn, out);
}